// Gauss_54133767799306
// MI455X (gfx1250) — compile-verified
//
#include <hip/hip_runtime.h>

typedef __attribute__((ext_vector_type(2))) float v2f;
typedef __attribute__((ext_vector_type(4))) float v4f;
typedef __attribute__((ext_vector_type(8))) float v8f;

#define CC 32          // channels == pw
#define PP 16          // patches
#define HH 128
#define WW 128
#define WSTRIDE 36     // padded LDS row stride (floats): 36m mod 64 distinct -> conflict-free
#define WTILE (CC * WSTRIDE)          // floats per staged 32x32 weight matrix
#define ASTAGE_PER_WAVE (16 * WSTRIDE)

// One Gauss-Seidel sweep step i. out is updated in-place (patch i only);
// patches j<i are read from out, j>i from x.
__global__ __launch_bounds__(256)
void gauss_step_kernel(const float* __restrict__ x,
                       const float* __restrict__ bsrc,
                       const float* __restrict__ Aw,
                       const float* __restrict__ Ab,
                       float* out, int i)
{
    extern __shared__ float smem[];
    float* Wlds   = smem;                      // 16 * 32 * WSTRIDE floats
    float* Astage = smem + PP * WTILE;         // 8 waves * 16 * WSTRIDE floats

    const int tid   = threadIdx.x;
    const int lane  = tid & 31;
    const int wave  = tid >> 5;
    const int mrow  = lane & 15;   // M (or N) index within 16-wide tile
    const int khalf = lane >> 4;   // selects K pair within 4-wide chunk

    float* Awave = Astage + wave * ASTAGE_PER_WAVE;

    // ---- Stage A_w[i] : 16 matrices of 32x32 fp32 into padded LDS ----
    {
        const float* Wsrc = Aw + (size_t)i * (PP * CC * CC);
        // 16*32*8 = 4096 float4 chunks, 256 threads
        for (int t = tid; t < PP * CC * (CC / 4); t += 256) {
            int k4 = t & 7;
            int n  = (t >> 3) & 31;
            int j  = t >> 8;
            v4f w = *(const v4f*)(Wsrc + ((size_t)(j * CC + n) * CC) + k4 * 4);
            float* dst = Wlds + j * WTILE + n * WSTRIDE + k4 * 4;
            dst[0] = w.x; dst[1] = w.y; dst[2] = w.z; dst[3] = w.w;
        }
    }

    // ---- Bias terms: diagonal bias and branch-free bias sum over j != i ----
    // bs = (sum over ALL j of Ab[i,j,n]) - Ab[i,i,n]
    const float* Abi = Ab + (size_t)i * (PP * CC);
    const float bd0 = Abi[i * CC + mrow];
    const float bd1 = Abi[i * CC + 16 + mrow];
    float bs0 = -bd0, bs1 = -bd1;
    #pragma unroll
    for (int j = 0; j < PP; ++j) {
        bs0 += Abi[j * CC + mrow];
        bs1 += Abi[j * CC + 16 + mrow];
    }

    __syncthreads();

    // ---- This wave's 16-row tile ----
    const int rbase = (blockIdx.x * 8 + wave) * 16;   // rows = (img<<5)|h_local
    const int ih = i >> 2, iw = i & 3;

    v8f acc0 = {};  // N = 0..15
    v8f acc1 = {};  // N = 16..31

    for (int j = 0; j < PP; ++j) {
        if (j == i) continue;
        const int jh = j >> 2, jw = j & 3;
        const float* src = (j < i) ? (const float*)out : x;

        // Stage 16 rows x 32 floats (coalesced float4) into per-wave LDS
        #pragma unroll
        for (int p = 0; p < 4; ++p) {
            int f   = p * 32 + lane;      // float4 index 0..127
            int r   = f >> 3;             // local row 0..15
            int seg = f & 7;              // float4 within row
            int row = rbase + r;
            int img = row >> 5;
            int hl  = row & 31;
            const float* gp = src + ((size_t)(img * HH + jh * 32 + hl) * WW + jw * 32 + seg * 4);
            v4f v = *(const v4f*)gp;
            float* dp = Awave + r * WSTRIDE + seg * 4;
            dp[0] = v.x; dp[1] = v.y; dp[2] = v.z; dp[3] = v.w;
        }
        asm volatile("s_wait_dscnt 0" ::: "memory");

        const float* Wj = Wlds + j * WTILE;
        #pragma unroll
        for (int c = 0; c < 8; ++c) {
            int k = c * 4 + khalf * 2;
            // A frag: A[m][k],A[m][k+1] ; B frag: B[k][n]=W[n][k] pairs
            v2f a  = *(const v2f*)(Awave + mrow * WSTRIDE + k);
            v2f b0 = *(const v2f*)(Wj + mrow * WSTRIDE + k);
            v2f b1 = *(const v2f*)(Wj + (mrow + 16) * WSTRIDE + k);
            acc0 = __builtin_amdgcn_wmma_f32_16x16x4_f32(false, a, false, b0,
                                                         (short)0, acc0, false, false);
            acc1 = __builtin_amdgcn_wmma_f32_16x16x4_f32(false, a, false, b1,
                                                         (short)0, acc1, false, false);
        }
        asm volatile("s_wait_dscnt 0" ::: "memory"); // A reads done before restage
    }

    // ---- rhs = bp[i] - s ; reshape through LDS into A-layout ----
    float* R = Awave;  // reuse per-wave staging
    #pragma unroll
    for (int v = 0; v < 8; ++v) {
        int m   = v + 8 * khalf;          // D-layout row for this lane
        int row = rbase + m;
        int img = row >> 5;
        int hl  = row & 31;
        const float* bprow = bsrc + ((size_t)(img * HH + ih * 32 + hl) * WW + iw * 32);
        R[m * WSTRIDE + mrow]      = bprow[mrow]      - (acc0[v] + bs0);
        R[m * WSTRIDE + mrow + 16] = bprow[mrow + 16] - (acc1[v] + bs1);
    }
    asm volatile("s_wait_dscnt 0" ::: "memory");

    // ---- Diagonal solve: xi = rhs @ W[i,i]^T + b[i,i] ----
    v8f d0 = {};
    v8f d1 = {};
    const float* Wi = Wlds + i * WTILE;
    #pragma unroll
    for (int c = 0; c < 8; ++c) {
        int k = c * 4 + khalf * 2;
        v2f a  = *(const v2f*)(R + mrow * WSTRIDE + k);
        v2f b0 = *(const v2f*)(Wi + mrow * WSTRIDE + k);
        v2f b1 = *(const v2f*)(Wi + (mrow + 16) * WSTRIDE + k);
        d0 = __builtin_amdgcn_wmma_f32_16x16x4_f32(false, a, false, b0,
                                                   (short)0, d0, false, false);
        d1 = __builtin_amdgcn_wmma_f32_16x16x4_f32(false, a, false, b1,
                                                   (short)0, d1, false, false);
    }

    #pragma unroll
    for (int v = 0; v < 8; ++v) {
        int m   = v + 8 * khalf;
        int row = rbase + m;
        int img = row >> 5;
        int hl  = row & 31;
        float* orow = out + ((size_t)(img * HH + ih * 32 + hl) * WW + iw * 32);
        orow[mrow]      = d0[v] + bd0;
        orow[mrow + 16] = d1[v] + bd1;
    }
}

extern "C" void kernel_launch(void* const* d_in, const int* in_sizes, int n_in,
                              void* d_out, int out_size, void* d_ws, size_t ws_size,
                              hipStream_t stream) {
    const float* x  = (const float*)d_in[0];   // [64,32,128,128]
    const float* b  = (const float*)d_in[1];   // [64,32,128,128]
    const float* Aw = (const float*)d_in[2];   // [16,16,32,32]
    const float* Ab = (const float*)d_in[3];   // [16,16,32]
    float* out = (float*)d_out;                // [64,32,128,128]

    const int rows    = 64 * 32 * 32;          // 65536 (b,c,h_local) rows
    const int nblocks = rows / (8 * 16);       // 8 waves x 16 rows per block = 512
    const size_t smem = (size_t)(PP * WTILE + 8 * ASTAGE_PER_WAVE) * sizeof(float);

    for (int i = 0; i < PP; ++i) {
        gauss_step_kernel<<<nblocks, 256, smem, stream>>>(x, b, Aw, Ab, out, i);
    }
}